// CollaborativeRNNModel_14018773254184
// MI455X (gfx1250) — compile-verified
//
#include <hip/hip_runtime.h>
#include <hip/hip_bf16.h>

#define B_ 32
#define S_ 128
#define U_ 256
#define H_ 128
#define V_ 30001

typedef __attribute__((ext_vector_type(2))) float v2f;
typedef __attribute__((ext_vector_type(8))) float v8f;

__device__ __forceinline__ float sigmoidf_(float x) {
    return 1.0f / (1.0f + __expf(-x));
}

// ---------------------------------------------------------------------------
// Phase 1: per-batch-row sequential GRU scan. One block per batch row (rows are
// fully independent: state is indexed [b, users[b,t]]). The whole 256x128 f32
// user-state table (128 KB) lives in LDS (CDNA5 WGP has 320 KB), so gather and
// scatter never leave the WGP. Weights stream from L2 (192 MB -> resident).
// 256 threads: thread j owns output column j of the 2H-wide ru vector.
// ---------------------------------------------------------------------------
extern "C" __global__ __launch_bounds__(256)
void crnn_scan_kernel(const int* __restrict__ users, const int* __restrict__ items,
                      const float* __restrict__ h0,
                      const float* __restrict__ P_ru, const float* __restrict__ W_ru,
                      const float* __restrict__ b_ru,
                      const float* __restrict__ P_c, const float* __restrict__ W_c,
                      const float* __restrict__ b_c,
                      float* __restrict__ Hout) {
    extern __shared__ float smem[];
    float* st  = smem;                 // U*H = 32768 floats (128 KB)
    float* rub = st + U_ * H_;         // 2H  = 256 floats
    float* rhb = rub + 2 * H_;         // H   = 128 floats
    const int b   = blockIdx.x;
    const int tid = threadIdx.x;

    for (int idx = tid; idx < U_ * H_; idx += 256)
        st[idx] = h0[(size_t)b * U_ * H_ + idx];

    for (int t = 0; t < S_; ++t) {
        __syncthreads();
        const int u  = users[b * S_ + t];   // uniform -> scalar loads
        const int it = items[b * S_ + t];
        const float* hp = st + u * H_;

        // ru = sigmoid(P_ru[it] + h_prev @ W_ru + b_ru)   (thread = column j)
        float dot = 0.0f;
        #pragma unroll 8
        for (int k = 0; k < H_; ++k)                     // LDS broadcast * coalesced global
            dot = __fmaf_rn(hp[k], W_ru[k * (2 * H_) + tid], dot);
        rub[tid] = sigmoidf_(P_ru[(size_t)it * (2 * H_) + tid] + dot + b_ru[tid]);
        __syncthreads();

        if (tid < H_) rhb[tid] = rub[tid] * hp[tid];     // r * h_prev
        __syncthreads();

        if (tid < H_) {
            float dc = 0.0f;
            #pragma unroll 8
            for (int k = 0; k < H_; ++k)
                dc = __fmaf_rn(rhb[k], W_c[k * H_ + tid], dc);
            float c  = tanhf(P_c[(size_t)it * H_ + tid] + dc + b_c[tid]);
            float z  = rub[H_ + tid];
            float hv = hp[tid];
            float hn = z * hv + (1.0f - z) * c;
            st[u * H_ + tid] = hn;                                  // scatter
            Hout[((size_t)b * S_ + t) * H_ + tid] = hn;             // outs[b*S+t]
        }
    }
}

// ---------------------------------------------------------------------------
// Phase 2: logits = H[4096,128] @ ws[128,30001] in fp32 via V_WMMA_F32_16X16X4.
// 31.5 GFLOP, 491 MB output -> store-bandwidth bound (~21 us floor @ 23.3 TB/s);
// ws (15 MB) + H (2 MB) stay L2-resident so HBM traffic ~= output only.
// Block = 256 threads (8 waves), block tile 128(M) x 64(N); each wave owns a
// 32x32 sub-tile = 4 v8f accumulators, K=128 consumed in 32 x K=4 WMMA steps.
// A tile staged once in LDS with row stride 130 to dodge 64-bank conflicts.
// ---------------------------------------------------------------------------
extern "C" __global__ __launch_bounds__(256)
void crnn_logits_wmma_kernel(const float* __restrict__ A,   // [4096, 128]
                             const float* __restrict__ Wv,  // [128, V]
                             float* __restrict__ Cout) {    // [4096, V]
    extern __shared__ float smem[];        // 128 x 130 floats (A tile, padded)
    const int tid  = threadIdx.x;
    const int lane = tid & 31;
    const int wave = tid >> 5;
    const int wm = wave & 3;               // 4 wave-rows  * 32 = 128 M
    const int wn = wave >> 2;              // 2 wave-cols  * 32 =  64 N
    const int mBase = blockIdx.y * 128;
    const int nBase = blockIdx.x * 64;

    for (int idx = tid; idx < 128 * 128; idx += 256) {
        const int r = idx >> 7, c = idx & 127;
        smem[r * 130 + c] = A[(size_t)(mBase + r) * H_ + c];
    }
    __syncthreads();

    const int half  = lane >> 4;           // K-pair select (ISA f32 A/B layout)
    const int sub   = lane & 15;           // M (for A) / N (for B,D)
    const int arow0 = wm * 32 + sub;
    const int bcol0 = nBase + wn * 32 + sub;

    v8f acc[2][2] = {};

    for (int k = 0; k < H_; k += 4) {
        const int kk = k + (half << 1);    // lanes 0-15: K={k,k+1}; 16-31: K={k+2,k+3}
        v2f a[2], bb[2];
        #pragma unroll
        for (int ms = 0; ms < 2; ++ms)
            a[ms] = *(const v2f*)&smem[(arow0 + ms * 16) * 130 + kk];  // 8B aligned
        #pragma unroll
        for (int ns = 0; ns < 2; ++ns) {
            int col = bcol0 + ns * 16;
            if (col >= V_) col = V_ - 1;   // clamp at the N edge; stores are masked
            v2f bv;
            bv.x = Wv[(size_t)kk * V_ + col];
            bv.y = Wv[(size_t)(kk + 1) * V_ + col];
            bb[ns] = bv;
        }
        #pragma unroll
        for (int ms = 0; ms < 2; ++ms)
            #pragma unroll
            for (int ns = 0; ns < 2; ++ns)
                acc[ms][ns] = __builtin_amdgcn_wmma_f32_16x16x4_f32(
                    false, a[ms], false, bb[ns], (short)0, acc[ms][ns], false, false);
    }

    #pragma unroll
    for (int ms = 0; ms < 2; ++ms) {
        #pragma unroll
        for (int ns = 0; ns < 2; ++ns) {
            const int col = bcol0 + ns * 16;
            if (col < V_) {
                const int rbase = mBase + wm * 32 + ms * 16 + half * 8;
                #pragma unroll
                for (int i = 0; i < 8; ++i)    // D: M = vgpr + 8*half, N = lane&15
                    Cout[(size_t)(rbase + i) * V_ + col] = acc[ms][ns][i];
            }
        }
    }
}

extern "C" void kernel_launch(void* const* d_in, const int* in_sizes, int n_in,
                              void* d_out, int out_size, void* d_ws, size_t ws_size,
                              hipStream_t stream) {
    const int*   users = (const int*)  d_in[0];
    const int*   items = (const int*)  d_in[1];
    const float* h0    = (const float*)d_in[2];
    const float* P_ru  = (const float*)d_in[3];
    const float* W_ru  = (const float*)d_in[4];
    const float* b_ru  = (const float*)d_in[5];
    const float* P_c   = (const float*)d_in[6];
    const float* W_c   = (const float*)d_in[7];
    const float* b_c   = (const float*)d_in[8];
    const float* wsmat = (const float*)d_in[9];
    float* logits = (float*)d_out;
    float* Hout   = (float*)d_ws;      // [B*S, H] f32 = 2 MB scratch

    const size_t rnnLds = (size_t)(U_ * H_ + 2 * H_ + H_) * sizeof(float);  // ~132.6 KB
    hipLaunchKernelGGL(crnn_scan_kernel, dim3(B_), dim3(256), rnnLds, stream,
                       users, items, h0, P_ru, W_ru, b_ru, P_c, W_c, b_c, Hout);

    dim3 grid((V_ + 63) / 64, (B_ * S_) / 128);
    const size_t gemmLds = (size_t)(128 * 130) * sizeof(float);             // 66.6 KB
    hipLaunchKernelGGL(crnn_logits_wmma_kernel, grid, dim3(256), gemmLds, stream,
                       Hout, wsmat, logits);
}